// GATConv_65601330479115
// MI455X (gfx1250) — compile-verified
//
#include <hip/hip_runtime.h>
#include <math.h>

#define IN_F      256
#define HEADS     4
#define DOUT      32
#define HD        128   // HEADS*DOUT
#define NEG_SLOPE 0.2f

typedef float v2f __attribute__((ext_vector_type(2)));
typedef float v8f __attribute__((ext_vector_type(8)));

// ---------------------------------------------------------------------------
// float atomic-max via int-max / uint-min bit trick (works with -inf init)
// ---------------------------------------------------------------------------
__device__ __forceinline__ void atomicMaxF(float* addr, float val) {
    if (val >= 0.0f)
        atomicMax(reinterpret_cast<int*>(addr), __float_as_int(val));
    else
        atomicMin(reinterpret_cast<unsigned int*>(addr), __float_as_uint(val));
}

// ---------------------------------------------------------------------------
// K0: init rst = bias (broadcast per row), m = -inf, s = 0
// ---------------------------------------------------------------------------
__global__ void k_init(const float* __restrict__ bias, float* __restrict__ rst,
                       float* __restrict__ mmax, float* __restrict__ ssum, int n_total) {
    int idx = blockIdx.x * 256 + threadIdx.x;       // over N*HD
    if (idx >= n_total) return;
    int f = idx & (HD - 1);
    rst[idx] = bias[f];
    if (f < HEADS) {
        int n = idx >> 7;
        mmax[n * HEADS + f] = -INFINITY;
        ssum[n * HEADS + f] = 0.0f;
    }
}

// ---------------------------------------------------------------------------
// K1: h = feat @ fc_w.T   via V_WMMA_F32_16X16X4_F32
//   block = 128 threads (4 waves), one 16-row M-tile per block.
//   A tile (16x256 f32) staged in LDS, rows padded to 260 floats so the
//   16 lanes of a half-wave land on distinct banks (stride 260 % 64 == 4).
//   Each wave computes cols [32w, 32w+32): two 16x16 accumulators, so each
//   A fragment is reused twice and the LDS tile 8x across the block.
//   fc_w is 128 KB -> permanently L2-resident; B fragments read direct.
// A layout (ISA 7.12.2, 32-bit A 16x4): lane<16 -> M=lane, K={0,1};
//   lane>=16 -> M=lane-16, K={2,3}.  B mirrors with N=lane%16.
// C layout: VGPR r -> (M = r + 8*(lane>=16), N = lane%16).
// ---------------------------------------------------------------------------
__global__ __launch_bounds__(128) void k_gemm_wmma(const float* __restrict__ feat,
                                                   const float* __restrict__ fc_w,
                                                   float* __restrict__ hOut,
                                                   int n_nodes) {
    __shared__ float As[16][260];
    const int tid = threadIdx.x;
    const int m0  = blockIdx.x * 16;

    // cooperative coalesced load of the 16x256 A tile (1024 float4s)
    {
        const float4* f4 = reinterpret_cast<const float4*>(feat);
#pragma unroll
        for (int j = 0; j < 8; ++j) {
            int idx = tid + j * 128;            // 0..1023
            int row = idx >> 6;                 // 0..15
            int c4  = idx & 63;                 // 0..63
            int grow = m0 + row;
            if (grow >= n_nodes) grow = n_nodes - 1;   // clamp (N%16==0 normally)
            float4 v = f4[(size_t)grow * (IN_F / 4) + c4];
            *reinterpret_cast<float4*>(&As[row][c4 * 4]) = v;
        }
    }
    __syncthreads();

    const int lane = tid & 31;
    const int wv   = tid >> 5;       // wave 0..3
    const int half = lane >> 4;      // 0: K={0,1} / M=r ; 1: K={2,3} / M=r+8
    const int lr   = lane & 15;

    const int   colA = wv * 32 + lr;                       // first N-tile column
    const float* bA  = fc_w + (size_t)colA * IN_F;         // fc_w[col][*] == B^T row
    const float* bB  = bA + 16 * IN_F;                     // second N-tile (+16 cols)

    v8f acc0 = {};
    v8f acc1 = {};

#pragma unroll 4
    for (int k = 0; k < IN_F; k += 4) {
        const int kk = k + 2 * half;
        const float2 av  = *reinterpret_cast<const float2*>(&As[lr][kk]);
        const float2 b0v = *reinterpret_cast<const float2*>(bA + kk);
        const float2 b1v = *reinterpret_cast<const float2*>(bB + kk);
        v2f a;  a[0]  = av.x;  a[1]  = av.y;
        v2f b0; b0[0] = b0v.x; b0[1] = b0v.y;
        v2f b1; b1[0] = b1v.x; b1[1] = b1v.y;
        acc0 = __builtin_amdgcn_wmma_f32_16x16x4_f32(false, a, false, b0,
                                                     (short)0, acc0, false, false);
        acc1 = __builtin_amdgcn_wmma_f32_16x16x4_f32(false, a, false, b1,
                                                     (short)0, acc1, false, false);
    }

    // store: row = m0 + r + 8*half, col = wv*32 + lr (+16 for acc1)
    const int rowBase = m0 + 8 * half;
#pragma unroll
    for (int r = 0; r < 8; ++r) {
        int row = rowBase + r;
        if (row < n_nodes) {
            float* o = hOut + (size_t)row * HD + wv * 32 + lr;
            o[0]  = acc0[r];
            o[16] = acc1[r];
        }
    }
}

// ---------------------------------------------------------------------------
// K2: el[n,h] = <h[n,h,:], attn_l[h,:]> ; er likewise
// ---------------------------------------------------------------------------
__global__ void k_proj(const float* __restrict__ hmat,
                       const float* __restrict__ attn_l,
                       const float* __restrict__ attn_r,
                       float* __restrict__ el, float* __restrict__ er, int NH) {
    int idx = blockIdx.x * 256 + threadIdx.x;
    if (idx >= NH) return;
    int n  = idx >> 2;
    int hh = idx & 3;
    const float* hp = hmat + (size_t)n * HD + hh * DOUT;
    const float* al = attn_l + hh * DOUT;
    const float* ar = attn_r + hh * DOUT;
    float sl = 0.0f, sr = 0.0f;
#pragma unroll
    for (int d = 0; d < DOUT; ++d) {
        float v = hp[d];
        sl += v * al[d];
        sr += v * ar[d];
    }
    el[idx] = sl;
    er[idx] = sr;
}

// ---------------------------------------------------------------------------
// K3: e = LeakyReLU(el[src] + er[dst]); segment max into m[dst]
// ---------------------------------------------------------------------------
__device__ __forceinline__ float leaky(float x) {
    return x > 0.0f ? x : NEG_SLOPE * x;
}

__global__ void k_edge_max(const float* __restrict__ el, const float* __restrict__ er,
                           const int* __restrict__ src, const int* __restrict__ dst,
                           float* __restrict__ e_buf, float* __restrict__ mmax, int E) {
    int e = blockIdx.x * 256 + threadIdx.x;
    if (e >= E) return;
    int s = src[e], d = dst[e];
    float4 l = *reinterpret_cast<const float4*>(el + (size_t)s * HEADS);
    float4 r = *reinterpret_cast<const float4*>(er + (size_t)d * HEADS);
    float4 v;
    v.x = leaky(l.x + r.x);
    v.y = leaky(l.y + r.y);
    v.z = leaky(l.z + r.z);
    v.w = leaky(l.w + r.w);
    *reinterpret_cast<float4*>(e_buf + (size_t)e * HEADS) = v;
    float* mp = mmax + (size_t)d * HEADS;
    atomicMaxF(mp + 0, v.x);
    atomicMaxF(mp + 1, v.y);
    atomicMaxF(mp + 2, v.z);
    atomicMaxF(mp + 3, v.w);
}

// ---------------------------------------------------------------------------
// K4: ex = exp(e - m[dst]); segment sum into s[dst]; e_buf overwritten w/ ex
// ---------------------------------------------------------------------------
__global__ void k_exp_sum(const int* __restrict__ dst, float* __restrict__ e_buf,
                          const float* __restrict__ mmax, float* __restrict__ ssum, int E) {
    int e = blockIdx.x * 256 + threadIdx.x;
    if (e >= E) return;
    int d = dst[e];
    float4 v = *reinterpret_cast<const float4*>(e_buf + (size_t)e * HEADS);
    float4 m = *reinterpret_cast<const float4*>(mmax + (size_t)d * HEADS);
    if (!(m.x > -INFINITY)) m.x = 0.0f;   // mirror jnp.where(isfinite(m), m, 0)
    if (!(m.y > -INFINITY)) m.y = 0.0f;
    if (!(m.z > -INFINITY)) m.z = 0.0f;
    if (!(m.w > -INFINITY)) m.w = 0.0f;
    v.x = expf(v.x - m.x);
    v.y = expf(v.y - m.y);
    v.z = expf(v.z - m.z);
    v.w = expf(v.w - m.w);
    *reinterpret_cast<float4*>(e_buf + (size_t)e * HEADS) = v;
    float* sp = ssum + (size_t)d * HEADS;
    atomicAdd(sp + 0, v.x);
    atomicAdd(sp + 1, v.y);
    atomicAdd(sp + 2, v.z);
    atomicAdd(sp + 3, v.w);
}

// ---------------------------------------------------------------------------
// K5: rst[dst,h,:] += h[src,h,:] * (ex / s[dst,h])
//   128 consecutive threads per edge -> coalesced gather + coalesced atomics
// ---------------------------------------------------------------------------
__global__ void k_scatter(const float* __restrict__ hmat, const float* __restrict__ e_buf,
                          const float* __restrict__ ssum,
                          const int* __restrict__ src, const int* __restrict__ dst,
                          float* __restrict__ rst, int total) {
    int gid = blockIdx.x * 256 + threadIdx.x;   // over E*HD (204.8M < 2^31)
    if (gid >= total) return;
    int e  = gid >> 7;         // / HD
    int f  = gid & (HD - 1);
    int hh = f >> 5;           // / DOUT
    int s = src[e], d = dst[e];
    float alpha = e_buf[(size_t)e * HEADS + hh] / ssum[(size_t)d * HEADS + hh];
    float val   = hmat[(size_t)s * HD + f] * alpha;
    atomicAdd(rst + (size_t)d * HD + f, val);
}

// ---------------------------------------------------------------------------
// launcher
// ---------------------------------------------------------------------------
extern "C" void kernel_launch(void* const* d_in, const int* in_sizes, int n_in,
                              void* d_out, int out_size, void* d_ws, size_t ws_size,
                              hipStream_t stream) {
    const float* feat   = (const float*)d_in[0];
    const float* fc_w   = (const float*)d_in[1];
    const float* attn_l = (const float*)d_in[2];
    const float* attn_r = (const float*)d_in[3];
    const float* bias   = (const float*)d_in[4];
    const int*   src    = (const int*)d_in[5];
    const int*   dst    = (const int*)d_in[6];
    float*       rst    = (float*)d_out;

    const int N = in_sizes[0] / IN_F;   // 100000
    const int E = in_sizes[5];          // 1600000

    // workspace layout (floats)
    float* h_ws  = (float*)d_ws;                    // N*HD
    float* e_buf = h_ws  + (size_t)N * HD;          // E*HEADS
    float* el    = e_buf + (size_t)E * HEADS;       // N*HEADS
    float* er    = el    + (size_t)N * HEADS;       // N*HEADS
    float* mx    = er    + (size_t)N * HEADS;       // N*HEADS
    float* sm    = mx    + (size_t)N * HEADS;       // N*HEADS

    const int nHD = N * HD;
    k_init<<<(nHD + 255) / 256, 256, 0, stream>>>(bias, rst, mx, sm, nHD);

    k_gemm_wmma<<<(N + 15) / 16, 128, 0, stream>>>(feat, fc_w, h_ws, N);

    const int NH = N * HEADS;
    k_proj<<<(NH + 255) / 256, 256, 0, stream>>>(h_ws, attn_l, attn_r, el, er, NH);

    k_edge_max<<<(E + 255) / 256, 256, 0, stream>>>(el, er, src, dst, e_buf, mx, E);

    k_exp_sum<<<(E + 255) / 256, 256, 0, stream>>>(dst, e_buf, mx, sm, E);

    const int total = E * HD;
    k_scatter<<<(total + 255) / 256, 256, 0, stream>>>(h_ws, e_buf, sm, src, dst, rst, total);
}